// AdapterLayer_56040733278487
// MI455X (gfx1250) — compile-verified
//
#include <hip/hip_runtime.h>
#include <hip/hip_bf16.h>

#define NN   50000
#define EE   400000
#define HID  128
#define EDGEA 16
#define NRBF 32
#define STR  440   // padded LDS row stride for a 16x432 mlp_in tile
#define EPSV 1e-8f

typedef __attribute__((ext_vector_type(2))) float v2f;
typedef __attribute__((ext_vector_type(8))) float v8f;

__device__ __forceinline__ v8f wmma4(v2f a, v2f b, v8f c) {
  // D = A(16x4 f32) * B(4x16 f32) + C(16x16 f32), exact fp32 on the matrix pipe
  return __builtin_amdgcn_wmma_f32_16x16x4_f32(false, a, false, b, (short)0, c,
                                               false, false);
}

__device__ __forceinline__ v8f zero8() {
  v8f z = {0.f, 0.f, 0.f, 0.f, 0.f, 0.f, 0.f, 0.f};
  return z;
}

__device__ __forceinline__ float silu(float x) {
  return x / (1.0f + __expf(-x));
}

// ---------------------------------------------------------------------------
// Weight pair-swizzle: out pair i=(k2,n) = { W[2*k2][n], W[2*k2+1][n] } so the
// GEMM B-fragment (k0,k0+1)@column n is a single 8-byte load.
// ---------------------------------------------------------------------------
__global__ void swz_kernel(const float* __restrict__ W, float* __restrict__ out,
                           int K2, int Nc) {
  const int i = blockIdx.x * blockDim.x + threadIdx.x;
  if (i >= K2 * Nc) return;
  const int k2 = i / Nc, n = i % Nc;
  out[2 * i]     = W[(2 * k2) * Nc + n];
  out[2 * i + 1] = W[(2 * k2 + 1) * Nc + n];
}

// ---------------------------------------------------------------------------
// Kernel A: vecp[3N,384] = vec[3N,128] @ Wvp[128,384]
// one wave per PAIR of 16-row tiles (B fragments reused across both tiles)
// ---------------------------------------------------------------------------
__global__ void __launch_bounds__(256) vecp_kernel(const float* __restrict__ vec,
                                                   const float* __restrict__ Wvpp,
                                                   float* __restrict__ vecp) {
  const int lane = threadIdx.x & 31;
  const int wv   = threadIdx.x >> 5;
  const int tp   = blockIdx.x * 8 + wv;     // pair-tiles: ceil(9375/2)=4688
  if (tp >= 4688) return;
  const int m    = lane & 15;
  const int half = lane >> 4;
  const long r0  = (long)tp * 32;
  const long r1  = r0 + 16;
  const bool two = (r1 < (long)NN * 3);
  const long r1s = two ? r1 : r0;

  for (int g = 0; g < 3; ++g) {
    v8f acc0[8], acc1[8];
#pragma unroll
    for (int nt = 0; nt < 8; ++nt) { acc0[nt] = zero8(); acc1[nt] = zero8(); }
#pragma unroll 4
    for (int kk = 0; kk < 32; ++kk) {
      const int k0 = 4 * kk + 2 * half;                 // A/B k_local = j + 2*(lane/16)
      v2f a0 = *(const v2f*)(vec + (r0 + m) * HID + k0);
      v2f a1 = *(const v2f*)(vec + (r1s + m) * HID + k0);
      const v2f* bp = (const v2f*)Wvpp + (long)(2 * kk + half) * 384 + g * 128 + m;
#pragma unroll
      for (int nt = 0; nt < 8; ++nt) {
        v2f b = bp[nt * 16];
        acc0[nt] = wmma4(a0, b, acc0[nt]);
        acc1[nt] = wmma4(a1, b, acc1[nt]);
      }
    }
#pragma unroll
    for (int nt = 0; nt < 8; ++nt) {
      const int n = g * 128 + nt * 16 + m;              // C/D: N = lane%16
#pragma unroll
      for (int r = 0; r < 8; ++r) {                     // C/D: M = r + 8*(lane/16)
        vecp[(r0 + r + 8 * half) * 384 + n] = acc0[nt][r];
        if (two) vecp[(r1 + r + 8 * half) * 384 + n] = acc1[nt][r];
      }
    }
  }
}

// ---------------------------------------------------------------------------
// Kernel B: vec_dot = sum_v vec1*vec2 ; zero h_aggr / vec_aggr accumulators
// ---------------------------------------------------------------------------
__global__ void prep_kernel(const float* __restrict__ vecp,
                            float* __restrict__ vec_dot,
                            float* __restrict__ h_aggr,
                            float* __restrict__ vec_aggr) {
  const long i = (long)blockIdx.x * blockDim.x + threadIdx.x;
  if (i < (long)NN * 3 * HID) vec_aggr[i] = 0.0f;
  if (i < (long)NN * HID) {
    const int n = (int)(i / HID), hh = (int)(i % HID);
    const float* p = vecp + (long)n * 3 * 384 + hh;
    vec_dot[i] = p[0] * p[128] + p[384] * p[512] + p[768] * p[896];
    h_aggr[i]  = 0.0f;
  }
}

// ---------------------------------------------------------------------------
// Kernel C: fused per-edge MLP + scatter. One wave per block, 32 edges per
// wave as TWO 16-row WMMA tiles sharing every B fragment (b64 paired loads).
// ---------------------------------------------------------------------------
__global__ void __launch_bounds__(32) edge_kernel(
    const float* __restrict__ h,     const float* __restrict__ coord,
    const int*   __restrict__ eidx,  const float* __restrict__ eattr,
    const float* __restrict__ vecp,
    const float* __restrict__ W1p, const float* __restrict__ b1,
    const float* __restrict__ W2p, const float* __restrict__ b2,
    const float* __restrict__ W3p, const float* __restrict__ b3,
    float* __restrict__ h_aggr, float* __restrict__ vec_aggr) {
  __shared__ float sm[32 * STR];           // two stacked 16x432 tiles (56.3 KB)
  __shared__ int   srow[32], scol[32];
  __shared__ float sxij[96];

  const int lane   = threadIdx.x;          // 0..31
  const int tile32 = blockIdx.x;           // 12500 blocks x 32 edges, exact
  const int m    = lane & 15;
  const int half = lane >> 4;
  float* S0 = sm;                          // edge rows 0..15
  float* S1 = sm + 16 * STR;               // edge rows 16..31

  // ---- stage 0: all 32 lanes each own one edge ----
  {
    const int e = tile32 * 32 + lane;
    const int r = eidx[e];
    const int c = eidx[EE + e];
    srow[lane] = r; scol[lane] = c;
    const float x0 = coord[r * 3 + 0] - coord[c * 3 + 0];
    const float x1 = coord[r * 3 + 1] - coord[c * 3 + 1];
    const float x2 = coord[r * 3 + 2] - coord[c * 3 + 2];
    sxij[lane * 3 + 0] = x0; sxij[lane * 3 + 1] = x1; sxij[lane * 3 + 2] = x2;
    const float d = sqrtf(x0 * x0 + x1 * x1 + x2 * x2 + EPSV);
    const float delta = 5.0f / (NRBF - 1);
    const float coeff = -0.5f / (delta * delta);
#pragma unroll
    for (int j = 0; j < NRBF; ++j) {
      const float t = d - (float)j * delta;
      sm[lane * STR + 256 + j] = __expf(coeff * t * t);
    }
#pragma unroll
    for (int j = 0; j < EDGEA; ++j)
      sm[lane * STR + 416 + j] = eattr[(long)e * EDGEA + j];
  }
  __syncthreads();

  // ---- stage 1: gathers h[row], h[col] and the cross term ----
  for (int e = 0; e < 32; ++e) {
    const long rb = (long)srow[e] * HID;
    const long cb = (long)scol[e] * HID;
    const long rv = (long)srow[e] * 3 * 384;     // vec1 at +0
    const long cv = (long)scol[e] * 3 * 384;     // vec2 at +128
#pragma unroll
    for (int h0 = 0; h0 < 128; h0 += 32) {
      const int hh = h0 + lane;
      sm[e * STR + hh]       = h[rb + hh];
      sm[e * STR + 128 + hh] = h[cb + hh];
      const float s = vecp[rv + hh]       * vecp[cv + 128 + hh]
                    + vecp[rv + 384 + hh] * vecp[cv + 512 + hh]
                    + vecp[rv + 768 + hh] * vecp[cv + 896 + hh];
      sm[e * STR + 288 + hh] = s;
    }
  }
  __syncthreads();

  // ---- GEMM1: [32,432] @ W1[432,128], SiLU, stage into LDS cols 0..127 ----
  v8f acc0[8], acc1[8];
#pragma unroll
  for (int nt = 0; nt < 8; ++nt) { acc0[nt] = zero8(); acc1[nt] = zero8(); }
#pragma unroll 4
  for (int kk = 0; kk < 108; ++kk) {
    const int k0 = 4 * kk + 2 * half;
    v2f a0 = *(const v2f*)(S0 + m * STR + k0);
    v2f a1 = *(const v2f*)(S1 + m * STR + k0);
    const v2f* bp = (const v2f*)W1p + (long)(2 * kk + half) * HID + m;
#pragma unroll
    for (int nt = 0; nt < 8; ++nt) {
      v2f b = bp[nt * 16];
      acc0[nt] = wmma4(a0, b, acc0[nt]);
      acc1[nt] = wmma4(a1, b, acc1[nt]);
    }
  }
#pragma unroll
  for (int nt = 0; nt < 8; ++nt) {
    const int n = nt * 16 + m;
    const float bb = b1[n];
#pragma unroll
    for (int r = 0; r < 8; ++r) {
      S0[(r + 8 * half) * STR + n] = silu(acc0[nt][r] + bb);
      S1[(r + 8 * half) * STR + n] = silu(acc1[nt][r] + bb);
    }
  }
  __syncthreads();

  // ---- GEMM2: [32,128] @ W2[128,128], SiLU, stage into LDS cols 128..255 ----
#pragma unroll
  for (int nt = 0; nt < 8; ++nt) { acc0[nt] = zero8(); acc1[nt] = zero8(); }
#pragma unroll 4
  for (int kk = 0; kk < 32; ++kk) {
    const int k0 = 4 * kk + 2 * half;
    v2f a0 = *(const v2f*)(S0 + m * STR + k0);
    v2f a1 = *(const v2f*)(S1 + m * STR + k0);
    const v2f* bp = (const v2f*)W2p + (long)(2 * kk + half) * HID + m;
#pragma unroll
    for (int nt = 0; nt < 8; ++nt) {
      v2f b = bp[nt * 16];
      acc0[nt] = wmma4(a0, b, acc0[nt]);
      acc1[nt] = wmma4(a1, b, acc1[nt]);
    }
  }
#pragma unroll
  for (int nt = 0; nt < 8; ++nt) {
    const int n = nt * 16 + m;
    const float bb = b2[n];
#pragma unroll
    for (int r = 0; r < 8; ++r) {
      S0[(r + 8 * half) * STR + 128 + n] = silu(acc0[nt][r] + bb);
      S1[(r + 8 * half) * STR + 128 + n] = silu(acc1[nt][r] + bb);
    }
  }
  __syncthreads();

  // ---- GEMM3 group 0 (msg_h), both tiles share B: scatter into h_aggr ----
#pragma unroll
  for (int nt = 0; nt < 8; ++nt) { acc0[nt] = zero8(); acc1[nt] = zero8(); }
#pragma unroll 4
  for (int kk = 0; kk < 32; ++kk) {
    const int k0 = 4 * kk + 2 * half;
    v2f a0 = *(const v2f*)(S0 + m * STR + 128 + k0);
    v2f a1 = *(const v2f*)(S1 + m * STR + 128 + k0);
    const v2f* bp = (const v2f*)W3p + (long)(2 * kk + half) * 384 + m;
#pragma unroll
    for (int nt = 0; nt < 8; ++nt) {
      v2f b = bp[nt * 16];
      acc0[nt] = wmma4(a0, b, acc0[nt]);
      acc1[nt] = wmma4(a1, b, acc1[nt]);
    }
  }
#pragma unroll
  for (int nt = 0; nt < 8; ++nt) {
    const int n = nt * 16 + m;
    const float bb = b3[n];
#pragma unroll
    for (int r = 0; r < 8; ++r) {
      const int mm = r + 8 * half;
      atomicAdd(&h_aggr[(long)srow[mm] * HID + n],      acc0[nt][r] + bb);
      atomicAdd(&h_aggr[(long)srow[16 + mm] * HID + n], acc1[nt][r] + bb);
    }
  }

  // ---- GEMM3 groups 1&2 (msg_v, msg_x) fused, one tile at a time ----
  for (int t = 0; t < 2; ++t) {
    const float* St = t ? S1 : S0;
    const int eb = t * 16;
    v8f av[8], ax[8];
#pragma unroll
    for (int nt = 0; nt < 8; ++nt) { av[nt] = zero8(); ax[nt] = zero8(); }
#pragma unroll 2
    for (int kk = 0; kk < 32; ++kk) {
      const int k0 = 4 * kk + 2 * half;
      v2f a = *(const v2f*)(St + m * STR + 128 + k0);
      const v2f* bpv = (const v2f*)W3p + (long)(2 * kk + half) * 384 + 128 + m;
      const v2f* bpx = (const v2f*)W3p + (long)(2 * kk + half) * 384 + 256 + m;
#pragma unroll
      for (int nt = 0; nt < 8; ++nt) {
        v2f bv = bpv[nt * 16];
        av[nt] = wmma4(a, bv, av[nt]);
        v2f bx = bpx[nt * 16];
        ax[nt] = wmma4(a, bx, ax[nt]);
      }
    }
#pragma unroll
    for (int nt = 0; nt < 8; ++nt) {
      const int n = nt * 16 + m;
      const float bv = b3[128 + n];
      const float bx = b3[256 + n];
#pragma unroll
      for (int r = 0; r < 8; ++r) {
        const int mm = eb + r + 8 * half;
        const int node = srow[mm], cn = scol[mm];
        const float mv = av[nt][r] + bv;
        const float mx = ax[nt][r] + bx;
        const float* v3 = vecp + (long)cn * 3 * 384 + 256 + n;   // vec3 slice
        float* va = vec_aggr + (long)node * 3 * HID + n;
        atomicAdd(va,           v3[0]   * mv + sxij[mm * 3 + 0] * mx);
        atomicAdd(va + HID,     v3[384] * mv + sxij[mm * 3 + 1] * mx);
        atomicAdd(va + 2 * HID, v3[768] * mv + sxij[mm * 3 + 2] * mx);
      }
    }
  }
}

// ---------------------------------------------------------------------------
// Kernel D: o = h_aggr @ Wop + bop ; dh = vec_dot*o2 + o3 ; dvec = vec3*o1 + vec_aggr
// ---------------------------------------------------------------------------
__global__ void __launch_bounds__(256) out_kernel(
    const float* __restrict__ h_aggr, const float* __restrict__ vec_dot,
    const float* __restrict__ vecp,   const float* __restrict__ vec_aggr,
    const float* __restrict__ Wopp,   const float* __restrict__ bop,
    float* __restrict__ dh, float* __restrict__ dvec) {
  const int lane = threadIdx.x & 31;
  const int wv   = threadIdx.x >> 5;
  const int tile = blockIdx.x * 8 + wv;
  if (tile >= NN / 16) return;            // 3125 tiles
  const int m    = lane & 15;
  const int half = lane >> 4;
  const long nb  = (long)tile * 16;

  v8f o2s[8];
#pragma unroll
  for (int g = 0; g < 3; ++g) {
    v8f acc[8];
#pragma unroll
    for (int nt = 0; nt < 8; ++nt) acc[nt] = zero8();
#pragma unroll 4
    for (int kk = 0; kk < 32; ++kk) {
      const int k0 = 4 * kk + 2 * half;
      v2f a = *(const v2f*)(h_aggr + (nb + m) * HID + k0);
      const v2f* bp = (const v2f*)Wopp + (long)(2 * kk + half) * 384 + g * 128 + m;
#pragma unroll
      for (int nt = 0; nt < 8; ++nt) {
        v2f b = bp[nt * 16];
        acc[nt] = wmma4(a, b, acc[nt]);
      }
    }
    if (g == 0) {                          // o1 -> dvec
#pragma unroll
      for (int nt = 0; nt < 8; ++nt) {
        const int n = nt * 16 + m;
        const float bb = bop[n];
#pragma unroll
        for (int r = 0; r < 8; ++r) {
          const long node = nb + r + 8 * half;
          const float o1 = acc[nt][r] + bb;
          const float* v3 = vecp + node * 3 * 384 + 256 + n;
          const float* va = vec_aggr + node * 3 * HID + n;
          float* dv = dvec + node * 3 * HID + n;
          dv[0]       = v3[0]   * o1 + va[0];
          dv[HID]     = v3[384] * o1 + va[HID];
          dv[2 * HID] = v3[768] * o1 + va[2 * HID];
        }
      }
    } else if (g == 1) {                   // stash o2
#pragma unroll
      for (int nt = 0; nt < 8; ++nt) {
        const float bb = bop[128 + nt * 16 + m];
#pragma unroll
        for (int r = 0; r < 8; ++r) o2s[nt][r] = acc[nt][r] + bb;
      }
    } else {                               // o3 -> dh
#pragma unroll
      for (int nt = 0; nt < 8; ++nt) {
        const int n = nt * 16 + m;
        const float bb = bop[256 + n];
#pragma unroll
        for (int r = 0; r < 8; ++r) {
          const long node = nb + r + 8 * half;
          dh[node * HID + n] = vec_dot[node * HID + n] * o2s[nt][r]
                             + acc[nt][r] + bb;
        }
      }
    }
  }
}

// ---------------------------------------------------------------------------
extern "C" void kernel_launch(void* const* d_in, const int* in_sizes, int n_in,
                              void* d_out, int out_size, void* d_ws, size_t ws_size,
                              hipStream_t stream) {
  (void)in_sizes; (void)n_in; (void)out_size; (void)ws_size;
  const float* h     = (const float*)d_in[0];
  const float* vec   = (const float*)d_in[1];
  const float* coord = (const float*)d_in[2];
  const int*   eidx  = (const int*)  d_in[3];
  const float* eattr = (const float*)d_in[4];
  const float* Wvp   = (const float*)d_in[5];
  const float* W1    = (const float*)d_in[6];
  const float* b1    = (const float*)d_in[7];
  const float* W2    = (const float*)d_in[8];
  const float* b2    = (const float*)d_in[9];
  const float* W3    = (const float*)d_in[10];
  const float* b3    = (const float*)d_in[11];
  const float* Wop   = (const float*)d_in[12];
  const float* bop   = (const float*)d_in[13];

  float* ws       = (float*)d_ws;                       // ~359.3 MB total
  float* vecp     = ws;                                 // [N,3,384]
  float* vec_dot  = vecp    + (size_t)NN * 3 * 384;     // [N,128]
  float* h_aggr   = vec_dot + (size_t)NN * HID;         // [N,128]
  float* vec_aggr = h_aggr  + (size_t)NN * HID;         // [N,3,128]
  float* Wvpp     = vec_aggr + (size_t)NN * 3 * HID;    // paired weights below
  float* W1p      = Wvpp + (size_t)128 * 384;
  float* W2p      = W1p  + (size_t)432 * 128;
  float* W3p      = W2p  + (size_t)128 * 128;
  float* Wopp     = W3p  + (size_t)128 * 384;

  float* dh   = (float*)d_out;                          // [N,128]
  float* dvec = dh + (size_t)NN * HID;                  // [N,3,128]

  // pair-swizzle all weight matrices (k-pairs contiguous -> b64 B loads)
  swz_kernel<<<(64 * 384 + 255) / 256, 256, 0, stream>>>(Wvp, Wvpp, 64, 384);
  swz_kernel<<<(216 * 128 + 255) / 256, 256, 0, stream>>>(W1, W1p, 216, 128);
  swz_kernel<<<(64 * 128 + 255) / 256, 256, 0, stream>>>(W2, W2p, 64, 128);
  swz_kernel<<<(64 * 384 + 255) / 256, 256, 0, stream>>>(W3, W3p, 64, 384);
  swz_kernel<<<(64 * 384 + 255) / 256, 256, 0, stream>>>(Wop, Wopp, 64, 384);

  vecp_kernel<<<(4688 + 7) / 8, 256, 0, stream>>>(vec, Wvpp, vecp);
  const long tot = (long)NN * 3 * HID;
  prep_kernel<<<(int)((tot + 255) / 256), 256, 0, stream>>>(vecp, vec_dot,
                                                            h_aggr, vec_aggr);
  edge_kernel<<<EE / 32, 32, 0, stream>>>(h, coord, eidx, eattr, vecp,
                                          W1p, b1, W2p, b2, W3p, b3,
                                          h_aggr, vec_aggr);
  out_kernel<<<(3125 + 7) / 8, 256, 0, stream>>>(h_aggr, vec_dot, vecp,
                                                 vec_aggr, Wopp, bop, dh, dvec);
}